// RGCNLayer_1932735283271
// MI455X (gfx1250) — compile-verified
//
#include <hip/hip_runtime.h>
#include <cstdint>
#include <cstddef>

typedef __attribute__((ext_vector_type(2))) float v2f;
typedef __attribute__((ext_vector_type(8))) float v8f;
typedef __attribute__((ext_vector_type(4))) unsigned int u32x4;
typedef __attribute__((ext_vector_type(8))) int i32x8;
typedef __attribute__((ext_vector_type(4))) int i32x4;

#define NN    20000
#define NE    640000
#define INF   128
#define OUTF  128
#define NR    65
#define NB    65
#define BN_EPS 1e-5f

// ---- workspace layout (element offsets; float/int both 4 bytes) ----
#define HPRE_OFF  0
#define HPRE_SZ   (NN * OUTF)                 // 2,560,000 f32
#define CS_OFF    (HPRE_OFF + HPRE_SZ)        // colsum   128 f32
#define CSQ_OFF   (CS_OFF + OUTF)             // colsumsq 128 f32
#define CNT_OFF   (CSQ_OFF + OUTF)            // counts   65 i32
#define CUR_OFF   (CNT_OFF + NR)              // cursor   65 i32
#define ST_OFF    (CUR_OFF + NR)              // starts   66 i32
#define ZERO_CNT  (ST_OFF + NR + 1)           // zero everything above
#define W_OFF     ((ZERO_CNT + 3) & ~3)       // W: 65*128*128 f32 (16B aligned)
#define W_SZ      (NR * INF * OUTF)
#define SORT_OFF  (W_OFF + W_SZ)              // sortedIdx: 640,000 i32

__device__ __forceinline__ void atomic_add_f32(float* p, float v) {
  __hip_atomic_fetch_add(p, v, __ATOMIC_RELAXED, __HIP_MEMORY_SCOPE_AGENT);
}

// ---- stage a 128x128 f32 matrix (64 KB) from global into LDS ----
// Preferred path: Tensor Data Mover (one DMA descriptor, tracked by TENSORcnt).
// D# per CDNA5 ISA: group0 {count, lds_addr, global_addr,
// type=2}; group1 {data_size=4B, tensor_dim0=16384, tile_dim0=16384,
// tile_dim1=1, tensor_dim0_stride=16384}; groups 2/3 unused (<=2D tensor).
// This toolchain (clang-23 / therock-10.0) exposes the 6-arg builtin form.
__device__ __forceinline__ void stage_matrix_lds(const float* __restrict__ Wg,
                                                 float* Wl) {
#if __has_builtin(__builtin_amdgcn_tensor_load_to_lds) && \
    __has_builtin(__builtin_amdgcn_s_wait_tensorcnt)
  if (threadIdx.x < 32) {  // wave 0 (uniform branch, full EXEC) programs the TDM
    uint64_t ga = (uint64_t)(uintptr_t)Wg;
    u32x4 g0;
    g0[0] = 1u;                                  // count=1 (valid user D#)
    g0[1] = (uint32_t)(uintptr_t)Wl;             // lds_addr (LDS byte offset)
    g0[2] = (uint32_t)ga;                        // global_addr[31:0]
    g0[3] = (uint32_t)(ga >> 32) | (2u << 30);   // global_addr[56:32] | type=2
    i32x8 g1;
    g1[0] = 0x00020000;  // workgroup_mask=0 (not in cluster), data_size=4B
    g1[1] = 0x40000000;  // tensor_dim0 = 16384 (bits[31:16] = low half)
    g1[2] = 0x00010000;  // tensor_dim0 hi = 0, tensor_dim1 = 1
    g1[3] = 0x40000000;  // tensor_dim1 hi = 0, tile_dim0 = 16384
    g1[4] = 1;           // tile_dim1 = 1, tile_dim2 = 0
    g1[5] = 16384;       // tensor_dim0_stride (low 32)
    g1[6] = 0;           // stride hi / tensor_dim1_stride lo
    g1[7] = 0;
    i32x4 z4 = {0, 0, 0, 0};
    i32x8 z8 = {0, 0, 0, 0, 0, 0, 0, 0};
    __builtin_amdgcn_tensor_load_to_lds(g0, g1, z4, z4, z8, 0);
    __builtin_amdgcn_s_wait_tensorcnt(0);        // TENSORcnt == 0: tile in LDS
  }
  __syncthreads();
#else
  const float4* Wg4 = (const float4*)Wg;
  float4* Wl4 = (float4*)Wl;
  __builtin_prefetch(Wg4 + threadIdx.x, 0, 1);   // global_prefetch_b8
#pragma unroll
  for (int i = 0; i < (INF * OUTF / 4) / 128; ++i)
    Wl4[threadIdx.x + 128 * i] = Wg4[threadIdx.x + 128 * i];
  __syncthreads();
#endif
}

// ---------------- zero the accumulator / counter region ----------------
__global__ void k_zero(float* __restrict__ ws) {
  int stride = gridDim.x * blockDim.x;
  for (int i = blockIdx.x * blockDim.x + threadIdx.x; i < ZERO_CNT; i += stride)
    ws[i] = 0.0f;
}

// ---------------- W[r] = sum_b comp[r,b] * basis[b] ----------------
__global__ void k_compute_W(const float* __restrict__ basis,
                            const float* __restrict__ comp,
                            float* __restrict__ W) {
  __shared__ float c[NB];
  int r = blockIdx.x;
  if (threadIdx.x < NB) c[threadIdx.x] = comp[r * NB + threadIdx.x];
  __syncthreads();
  float* Wr = W + (size_t)r * (INF * OUTF);
  for (int idx = threadIdx.x; idx < INF * OUTF; idx += blockDim.x) {
    float acc = 0.0f;
    for (int b = 0; b < NB; ++b) acc += c[b] * basis[(size_t)b * (INF * OUTF) + idx];
    Wr[idx] = acc;
  }
}

// ---------------- counting sort of edges by relation ----------------
__global__ void k_hist(const int* __restrict__ etype, int* __restrict__ counts) {
  int stride = gridDim.x * blockDim.x;
  for (int e = blockIdx.x * blockDim.x + threadIdx.x; e < NE; e += stride)
    atomicAdd(&counts[etype[e]], 1);
}

__global__ void k_scan(const int* __restrict__ counts,
                       int* __restrict__ starts, int* __restrict__ cursor) {
  if (threadIdx.x == 0) {
    int acc = 0;
    for (int r = 0; r < NR; ++r) {
      starts[r] = acc;
      cursor[r] = acc;
      acc += counts[r];
    }
    starts[NR] = acc;
  }
}

__global__ void k_scatter(const int* __restrict__ etype,
                          int* __restrict__ cursor, int* __restrict__ sortedIdx) {
  int stride = gridDim.x * blockDim.x;
  for (int e = blockIdx.x * blockDim.x + threadIdx.x; e < NE; e += stride) {
    int pos = atomicAdd(&cursor[etype[e]], 1);
    sortedIdx[pos] = e;
  }
}

// ---------------- edge transform: h_pre[dst] += (x[src]*norm) @ W[etype] ----------------
// grid (NR, 32) x 128 threads (4 waves). Block stages W[r] (64KB) in LDS via TDM.
// Each wave handles tiles of 16 edges with V_WMMA_F32_16X16X4_F32.
__global__ void __launch_bounds__(128)
k_edge(const float* __restrict__ node_feats, const int* __restrict__ src,
       const int* __restrict__ dst, const float* __restrict__ norm,
       const float* __restrict__ W, const int* __restrict__ starts,
       const int* __restrict__ sortedIdx, float* __restrict__ h_pre) {
  __shared__ float Wl[INF * OUTF];  // 64 KB (WGP has 320 KB LDS)

  int r = blockIdx.x;
  int rs = starts[r], re = starts[r + 1];
  int cnt = re - rs;
  if (cnt == 0) return;  // block-uniform, before any barrier

  stage_matrix_lds(W + (size_t)r * (INF * OUTF), Wl);

  int wave = threadIdx.x >> 5;
  int lane = threadIdx.x & 31;
  int hi = lane >> 4;     // half-wave selector (K-pair for A/B fragments)
  int m  = lane & 15;     // row (edge in tile) for A; column for B/D
  int ntiles = (cnt + 15) >> 4;
  int slot = blockIdx.y * 4 + wave;  // 0..127 parallel wave slots per relation

  for (int t = slot; t < ntiles; t += 128) {
    int base = rs + t * 16;

    // ---- A fragment registers: lane holds row m, cols {4k+2hi, 4k+2hi+1} ----
    v2f areg[32];
    int posm = base + m;
    if (posm < re) {
      int e = sortedIdx[posm];
      const v2f* row2 = (const v2f*)(node_feats + (size_t)src[e] * INF);
      float nrm = norm[e];
#pragma unroll
      for (int kk = 0; kk < 32; ++kk) areg[kk] = row2[2 * kk + hi] * nrm;
    } else {
#pragma unroll
      for (int kk = 0; kk < 32; ++kk) { v2f z = {0.0f, 0.0f}; areg[kk] = z; }
    }

    // destination node for each of this lane's 8 output rows (M = v + 8*hi)
    int dstm[8];
#pragma unroll
    for (int v = 0; v < 8; ++v) {
      int pos = base + hi * 8 + v;
      dstm[v] = (pos < re) ? dst[sortedIdx[pos]] : -1;
    }

    // ---- 8 column tiles x 32 K-steps of f32 WMMA ----
    for (int nt = 0; nt < 8; ++nt) {
      v8f acc = {};
#pragma unroll
      for (int kk = 0; kk < 32; ++kk) {
        int krow = kk * 4 + hi * 2;
        v2f b;
        b[0] = Wl[(krow + 0) * OUTF + nt * 16 + m];
        b[1] = Wl[(krow + 1) * OUTF + nt * 16 + m];
        acc = __builtin_amdgcn_wmma_f32_16x16x4_f32(
            false, areg[kk], false, b, (short)0, acc, false, false);
      }
#pragma unroll
      for (int v = 0; v < 8; ++v) {
        if (dstm[v] >= 0)
          atomic_add_f32(&h_pre[(size_t)dstm[v] * OUTF + nt * 16 + m], acc[v]);
      }
    }
  }
}

// ---------------- residual GEMM + fuse + column stats ----------------
// out = relu(h_pre + h_bias) + relu(x @ W_res + b_res); accumulate col sums.
__global__ void __launch_bounds__(128)
k_res(const float* __restrict__ node_feats, const float* __restrict__ W_res,
      const float* __restrict__ b_res, const float* __restrict__ h_bias,
      const float* __restrict__ h_pre, float* __restrict__ out,
      float* __restrict__ colsum, float* __restrict__ colsumsq) {
  __shared__ float Wl[INF * OUTF];  // 64 KB

  stage_matrix_lds(W_res, Wl);

  int wave = threadIdx.x >> 5;
  int lane = threadIdx.x & 31;
  int hi = lane >> 4;
  int m  = lane & 15;
  int slot = blockIdx.x * 4 + wave;
  int stride = gridDim.x * 4;
  const int NTILES = NN / 16;  // 1250

  for (int t = slot; t < NTILES; t += stride) {
    int base = t * 16;

    v2f areg[32];
    {
      const v2f* row2 = (const v2f*)(node_feats + (size_t)(base + m) * INF);
#pragma unroll
      for (int kk = 0; kk < 32; ++kk) areg[kk] = row2[2 * kk + hi];
    }

    for (int nt = 0; nt < 8; ++nt) {
      v8f acc = {};
#pragma unroll
      for (int kk = 0; kk < 32; ++kk) {
        int krow = kk * 4 + hi * 2;
        v2f b;
        b[0] = Wl[(krow + 0) * OUTF + nt * 16 + m];
        b[1] = Wl[(krow + 1) * OUTF + nt * 16 + m];
        acc = __builtin_amdgcn_wmma_f32_16x16x4_f32(
            false, areg[kk], false, b, (short)0, acc, false, false);
      }
      int col = nt * 16 + m;
      float bres = b_res[col];
      float hb = h_bias[col];
      float psum = 0.0f, psq = 0.0f;
#pragma unroll
      for (int v = 0; v < 8; ++v) {
        int row = base + hi * 8 + v;
        float conv = h_pre[(size_t)row * OUTF + col] + hb;
        conv = conv > 0.0f ? conv : 0.0f;
        float resv = acc[v] + bres;
        resv = resv > 0.0f ? resv : 0.0f;
        float h = conv + resv;
        out[(size_t)row * OUTF + col] = h;
        psum += h;
        psq += h * h;
      }
      atomic_add_f32(&colsum[col], psum);
      atomic_add_f32(&colsumsq[col], psq);
    }
  }
}

// ---------------- BN: fold stats into per-column scale/shift ----------------
__global__ void k_bn_prep(float* __restrict__ colsum, float* __restrict__ colsumsq,
                          const float* __restrict__ gamma, const float* __restrict__ beta) {
  int c = threadIdx.x;
  if (c < OUTF) {
    float mean = colsum[c] * (1.0f / (float)NN);
    float var = colsumsq[c] * (1.0f / (float)NN) - mean * mean;
    float s = gamma[c] * rsqrtf(var + BN_EPS);
    colsum[c] = s;                     // scale
    colsumsq[c] = beta[c] - mean * s;  // shift
  }
}

__global__ void k_bn(float* __restrict__ out, const float* __restrict__ scale,
                     const float* __restrict__ shift) {
  const float4* s4 = (const float4*)scale;
  const float4* sh4 = (const float4*)shift;
  float4* o4 = (float4*)out;
  int total = NN * OUTF / 4;
  int stride = gridDim.x * blockDim.x;
  for (int i = blockIdx.x * blockDim.x + threadIdx.x; i < total; i += stride) {
    int c4 = i & (OUTF / 4 - 1);
    float4 v = o4[i];
    float4 sc = s4[c4], sf = sh4[c4];
    v.x = v.x * sc.x + sf.x;
    v.y = v.y * sc.y + sf.y;
    v.z = v.z * sc.z + sf.z;
    v.w = v.w * sc.w + sf.w;
    o4[i] = v;
  }
}

extern "C" void kernel_launch(void* const* d_in, const int* in_sizes, int n_in,
                              void* d_out, int out_size, void* d_ws, size_t ws_size,
                              hipStream_t stream) {
  const float* node_feats = (const float*)d_in[0];
  const int*   src        = (const int*)d_in[1];
  const int*   dst        = (const int*)d_in[2];
  const int*   etype      = (const int*)d_in[3];
  const float* norm       = (const float*)d_in[4];
  const float* basis      = (const float*)d_in[5];
  const float* comp       = (const float*)d_in[6];
  const float* h_bias     = (const float*)d_in[7];
  const float* W_res      = (const float*)d_in[8];
  const float* b_res      = (const float*)d_in[9];
  const float* gamma      = (const float*)d_in[10];
  const float* beta       = (const float*)d_in[11];
  float* out = (float*)d_out;

  float* wsf = (float*)d_ws;
  int*   wsi = (int*)d_ws;
  float* h_pre     = wsf + HPRE_OFF;
  float* colsum    = wsf + CS_OFF;
  float* colsumsq  = wsf + CSQ_OFF;
  int*   counts    = wsi + CNT_OFF;
  int*   cursor    = wsi + CUR_OFF;
  int*   starts    = wsi + ST_OFF;
  float* W         = wsf + W_OFF;
  int*   sortedIdx = wsi + SORT_OFF;

  k_zero<<<2048, 256, 0, stream>>>(wsf);
  k_compute_W<<<NR, 256, 0, stream>>>(basis, comp, W);
  k_hist<<<1024, 256, 0, stream>>>(etype, counts);
  k_scan<<<1, 64, 0, stream>>>(counts, starts, cursor);
  k_scatter<<<1024, 256, 0, stream>>>(etype, cursor, sortedIdx);
  k_edge<<<dim3(NR, 32), 128, 0, stream>>>(node_feats, src, dst, norm, W,
                                           starts, sortedIdx, h_pre);
  k_res<<<313, 128, 0, stream>>>(node_feats, W_res, b_res, h_bias, h_pre,
                                 out, colsum, colsumsq);
  k_bn_prep<<<1, 128, 0, stream>>>(colsum, colsumsq, gamma, beta);
  k_bn<<<1280, 256, 0, stream>>>(out, colsum, colsumsq);
}